// PrecisionFocusedLoss_3418793968004
// MI455X (gfx1250) — compile-verified
//
#include <hip/hip_runtime.h>

typedef __attribute__((ext_vector_type(2))) float v2f;
typedef __attribute__((ext_vector_type(8))) float v8f;

#define BLOCKS  2048
#define THREADS 256

// ---- per-sample weighted cross-entropy (2 classes) ----------------------
__device__ __forceinline__ float sample_loss(float l0, float l1, int t) {
    float m   = fmaxf(l0, l1);
    float d   = fabsf(l0 - l1);
    // logsumexp(l0,l1) = m + log(1 + exp(-|l0-l1|))   (stable; arg in [1,2])
    float lse = m + __logf(1.0f + __expf(-d));
    float ce  = lse - (t ? l1 : l0);
    int   p   = (l1 > l0) ? 1 : 0;              // argmax, ties -> class 0
    float mult = 1.3f;                          // 1 + 3*0.1
    if (t == 1 && p == 0)      mult = 4.0f;     // false negative: 1 + 3*1
    else if (t == 0 && p == 1) mult = 16.0f;    // false positive: 1 + 3*5
    return ce * mult;
}

// ---- wave32 sum via V_WMMA_F32_16X16X4_F32 ------------------------------
// A (16x4 f32): VGPR0 = {lanes 0-15 -> K=0, lanes 16-31 -> K=2}, VGPR1 = 0.
// B = all ones  =>  D[m][n] = rowsum(A[m]).  In lane L<16, d[0..7] hold
// D[0..7][L]; in lanes 16-31, d[0..7] hold D[8..15][.].  Sum the 8 regs,
// then fold the two half-waves with one shfl_xor(16).
__device__ __forceinline__ float wave_sum_wmma(float acc) {
    v2f a; a[0] = acc;  a[1] = 0.0f;
    v2f b; b[0] = 1.0f; b[1] = 1.0f;
    v8f c = {};
    v8f dm = __builtin_amdgcn_wmma_f32_16x16x4_f32(
        /*neg_a=*/false, a, /*neg_b=*/false, b,
        /*c_mod=*/(short)0, c, /*reuse_a=*/false, /*reuse_b=*/false);
    float s = 0.0f;
#pragma unroll
    for (int i = 0; i < 8; ++i) s += dm[i];
    s += __shfl_xor(s, 16);
    return s;                                   // wave total in every lane
}

__device__ __forceinline__ float block_sum(float acc) {
    __shared__ float sm[THREADS / 32];
    float w    = wave_sum_wmma(acc);
    int   lane = threadIdx.x & 31;
    int   wid  = threadIdx.x >> 5;
    if (lane == 0) sm[wid] = w;
    __syncthreads();
    float t = 0.0f;
    if (threadIdx.x == 0) {
#pragma unroll
        for (int i = 0; i < THREADS / 32; ++i) t += sm[i];
    }
    return t;                                   // valid in thread 0 only
}

// ---- pass 1: streaming CE + per-block partial sums ----------------------
__global__ __launch_bounds__(THREADS)
void ce_partial_kernel(const float* __restrict__ logits,
                       const int*   __restrict__ targets,
                       float*       __restrict__ partials,
                       int n) {
    const long tid      = (long)blockIdx.x * blockDim.x + threadIdx.x;
    const long nthreads = (long)gridDim.x * blockDim.x;

    float acc = 0.0f;
    for (long base = tid * 4; base < n; base += nthreads * 4) {
        if (base + 3 < n) {
            // 2 x float4 = logits for samples base..base+3 (32B, aligned)
            float4 La = *(const float4*)(logits + base * 2);
            float4 Lb = *(const float4*)(logits + base * 2 + 4);
            int4   T  = *(const int4*)(targets + base);
            acc += sample_loss(La.x, La.y, T.x);
            acc += sample_loss(La.z, La.w, T.y);
            acc += sample_loss(Lb.x, Lb.y, T.z);
            acc += sample_loss(Lb.z, Lb.w, T.w);
        } else {
            for (long i = base; i < n; ++i)
                acc += sample_loss(logits[2 * i], logits[2 * i + 1], targets[i]);
        }
    }

    float bs = block_sum(acc);
    if (threadIdx.x == 0) partials[blockIdx.x] = bs;
}

// ---- pass 2: reduce partials, write mean --------------------------------
__global__ __launch_bounds__(THREADS)
void ce_final_kernel(const float* __restrict__ partials, int nparts,
                     float* __restrict__ out, float inv_n) {
    float acc = 0.0f;
    for (int i = threadIdx.x; i < nparts; i += blockDim.x)
        acc += partials[i];
    float t = block_sum(acc);
    if (threadIdx.x == 0) out[0] = t * inv_n;
}

extern "C" void kernel_launch(void* const* d_in, const int* in_sizes, int n_in,
                              void* d_out, int out_size, void* d_ws, size_t ws_size,
                              hipStream_t stream) {
    const float* logits  = (const float*)d_in[0];
    const int*   targets = (const int*)d_in[1];
    float*       out     = (float*)d_out;
    float*       partials = (float*)d_ws;       // BLOCKS floats of scratch

    const int n = in_sizes[1];                  // number of samples (B)

    ce_partial_kernel<<<BLOCKS, THREADS, 0, stream>>>(logits, targets, partials, n);
    ce_final_kernel<<<1, THREADS, 0, stream>>>(partials, BLOCKS, out, 1.0f / (float)n);
}